// GCN_1_83631603187959
// MI455X (gfx1250) — compile-verified
//
#include <hip/hip_runtime.h>
#include <hip/hip_bf16.h>
#include <stdint.h>

#define N_NODES 100000
#define N_EDGES 1600000
#define IN_DIM  512
#define OUT_DIM 64

typedef __attribute__((ext_vector_type(2))) float    v2f;
typedef __attribute__((ext_vector_type(8))) float    v8f;
typedef __attribute__((ext_vector_type(4))) uint32_t v4u;
typedef __attribute__((ext_vector_type(8))) int      v8i;
typedef __attribute__((ext_vector_type(4))) int      v4i;

#if defined(__has_builtin)
#if __has_builtin(__builtin_amdgcn_tensor_load_to_lds) && \
    __has_builtin(__builtin_amdgcn_s_wait_tensorcnt)
#define HAVE_TDM 1
#endif
#endif

// ---------------- degree / norm ----------------

__global__ void k_init_deg(uint32_t* __restrict__ deg) {
    int i = blockIdx.x * blockDim.x + threadIdx.x;
    if (i < N_NODES) deg[i] = 1u;  // self-loop contributes 1
}

__global__ void k_degree(const long long* __restrict__ ei, uint32_t* __restrict__ deg) {
    int e = blockIdx.x * blockDim.x + threadIdx.x;
    if (e < N_EDGES) {
        int d = (int)ei[(size_t)N_EDGES + e];  // dst row
        atomicAdd(&deg[d], 1u);
    }
}

__global__ void k_dinv(const uint32_t* __restrict__ deg, float* __restrict__ dinv) {
    int i = blockIdx.x * blockDim.x + threadIdx.x;
    if (i < N_NODES) dinv[i] = rsqrtf((float)deg[i]);  // deg >= 1 always
}

// ---------------- GEMM: h = x @ W via V_WMMA_F32_16X16X4_F32 ----------------
// W (512x64 f32 = 128 KB) staged into LDS once per block via the Tensor Data
// Mover (TDM); wave -> one 16x16 tile of h; block (8 waves) -> 32 rows x 64 cols.

__global__ void __launch_bounds__(256)
k_gemm(const float* __restrict__ x, const float* __restrict__ W, float* __restrict__ h) {
    __shared__ float sW[IN_DIM * OUT_DIM];   // 128 KB of 320 KB WGP LDS

#if HAVE_TDM
    if (threadIdx.x < 32) {                  // wave 0 issues the TDM descriptor
        const uint64_t ga = (uint64_t)(uintptr_t)W;
        // D# group 0: count=1, lds_addr=0 (sW is the only LDS object),
        // global_addr, type=2 ("image")
        v4u g0;
        g0.x = 1u;                                   // count = 1 valid descriptor
        g0.y = 0u;                                   // lds_addr (bytes)
        g0.z = (uint32_t)(ga & 0xffffffffu);         // global_addr[31:0]
        g0.w = (uint32_t)((ga >> 32) & 0x01ffffffu)  // global_addr[56:32]
             | (2u << 30);                           // type = 2
        // D# group 1: data_size=4B (code 2); tensor 64 x 512, stride0 = 64;
        // tile = whole tensor (tile_dim0=64, tile_dim1=512)
        v8i g1;
        g1[0] = (int)(2u << 16);                     // data_size=2 (4 bytes)
        g1[1] = (int)((64u & 0xffffu) << 16);        // tensor_dim0[15:0]
        g1[2] = (int)((512u & 0xffffu) << 16);       // dim0 hi | tensor_dim1[15:0]
        g1[3] = (int)((64u & 0xffffu) << 16);        // dim1 hi | tile_dim0
        g1[4] = (int)512u;                           // tile_dim1 | tile_dim2=0
        g1[5] = (int)64u;                            // tensor_dim0_stride[31:0]
        g1[6] = 0;
        g1[7] = 0;
        v4i gz4 = {0, 0, 0, 0};                      // groups 2/3 unused (2D)
        v8i gz8 = {0, 0, 0, 0, 0, 0, 0, 0};         // 5th group (clang-23 6-arg form)
        __builtin_amdgcn_tensor_load_to_lds(g0, g1, gz4, gz4, gz8, 0);
        __builtin_amdgcn_s_wait_tensorcnt(0);        // TENSORcnt -> 0
    }
    // The TDM builtin doesn't reference sW, so escape the pointer and clobber
    // memory: forces LLVM to keep the LDS allocation and treat sW as written
    // (otherwise the B-fragment ds_loads get folded to undef and LDS is DCE'd).
    {
        void* sWp = (void*)sW;
        asm volatile("" : "+v"(sWp) : : "memory");
    }
    __syncthreads();
#else
    // Fallback: cooperative vector copy of W into LDS
    for (int i = threadIdx.x; i < (IN_DIM * OUT_DIM) / 4; i += 256) {
        ((float4*)sW)[i] = ((const float4*)W)[i];
    }
    __syncthreads();
#endif

    const int wave = threadIdx.x >> 5;          // 0..7
    const int lane = threadIdx.x & 31;
    const int half = lane >> 4;                 // 0: lanes 0-15, 1: lanes 16-31
    const int l    = lane & 15;
    const int m0 = blockIdx.x * 32 + (wave >> 2) * 16;  // row tile
    const int n0 = (wave & 3) * 16;                     // col tile

    const float* __restrict__ arow = x + (size_t)(m0 + l) * IN_DIM;
    const float* swcol = sW + n0 + l;

    v8f acc = {};
    #pragma unroll 4
    for (int k = 0; k < IN_DIM; k += 4) {
        const int ka = k + 2 * half;
        // A 16x4 f32: lanes 0-15 -> K = k, k+1 ; lanes 16-31 -> K = k+2, k+3
        v2f a = *(const v2f*)(arow + ka);
        // B 4x16 f32 from LDS: VGPR0 -> K=k (lo lanes) / k+2 (hi lanes); VGPR1 -> K+1
        v2f bf;
        bf.x = swcol[ka * OUT_DIM];
        bf.y = swcol[(ka + 1) * OUT_DIM];
        acc = __builtin_amdgcn_wmma_f32_16x16x4_f32(
            /*neg_a=*/false, a, /*neg_b=*/false, bf,
            /*c_mod=*/(short)0, acc, /*reuse_a=*/false, /*reuse_b=*/false);
    }

    // C/D: VGPR r holds row m0+r (lanes 0-15) / m0+8+r (lanes 16-31), col n0+l
    float* __restrict__ hrow = h + (size_t)(m0 + 8 * half) * OUT_DIM + n0 + l;
    #pragma unroll
    for (int r = 0; r < 8; ++r) hrow[(size_t)r * OUT_DIM] = acc[r];
}

// ---------------- out = b + self-loop message (initializes d_out) ----------------

__global__ void k_self(const float* __restrict__ h, const float* __restrict__ dinv,
                       const float* __restrict__ b, float* __restrict__ out) {
    int t = blockIdx.x * blockDim.x + threadIdx.x;
    if (t < N_NODES * OUT_DIM) {
        int i = t >> 6, j = t & 63;
        float di = dinv[i];
        out[t] = b[j] + h[t] * di * di;
    }
}

// ---------------- edge scatter: wave per edge, lane covers 2 columns ----------------

__global__ void __launch_bounds__(256)
k_scatter(const long long* __restrict__ ei, const float* __restrict__ h,
          const float* __restrict__ dinv, float* __restrict__ out) {
    const int gwave = (int)((blockIdx.x * (unsigned)blockDim.x + threadIdx.x) >> 5);
    const int lane  = threadIdx.x & 31;
    if (gwave >= N_EDGES) return;
    const long long s = ei[gwave];
    const long long d = ei[(size_t)N_EDGES + gwave];
    const float norm = dinv[s] * dinv[d];
    const float* __restrict__ hs = h + (size_t)s * OUT_DIM;
    float* __restrict__ od = out + (size_t)d * OUT_DIM;
    const int c = lane * 2;
    unsafeAtomicAdd(&od[c + 0], hs[c + 0] * norm);  // global_atomic_add_f32, L2-resident
    unsafeAtomicAdd(&od[c + 1], hs[c + 1] * norm);
}

// ---------------- row-wise log_softmax, wave per row ----------------

__global__ void __launch_bounds__(256)
k_logsoftmax(float* __restrict__ out) {
    const int row  = (int)((blockIdx.x * (unsigned)blockDim.x + threadIdx.x) >> 5);
    const int lane = threadIdx.x & 31;
    if (row >= N_NODES) return;
    float* __restrict__ o = out + (size_t)row * OUT_DIM;
    float v0 = o[lane], v1 = o[lane + 32];
    float m = fmaxf(v0, v1);
    #pragma unroll
    for (int off = 16; off > 0; off >>= 1) m = fmaxf(m, __shfl_xor(m, off, 32));
    float s = __expf(v0 - m) + __expf(v1 - m);
    #pragma unroll
    for (int off = 16; off > 0; off >>= 1) s += __shfl_xor(s, off, 32);
    const float lse = m + __logf(s);
    o[lane]      = v0 - lse;
    o[lane + 32] = v1 - lse;
}

// ---------------- launcher ----------------

extern "C" void kernel_launch(void* const* d_in, const int* in_sizes, int n_in,
                              void* d_out, int out_size, void* d_ws, size_t ws_size,
                              hipStream_t stream) {
    const float*     x  = (const float*)d_in[0];
    const long long* ei = (const long long*)d_in[1];   // int64 [2, E]
    const float*     W  = (const float*)d_in[2];
    const float*     b  = (const float*)d_in[3];
    float* out = (float*)d_out;

    char* ws = (char*)d_ws;
    uint32_t* deg  = (uint32_t*)ws;                                  // N u32
    float*    dinv = (float*)(ws + 4ull * N_NODES);                  // N f32
    float*    h    = (float*)(ws + 8ull * N_NODES);                  // N x 64 f32 (25.6 MB)

    k_init_deg<<<(N_NODES + 255) / 256, 256, 0, stream>>>(deg);
    k_degree  <<<(N_EDGES + 255) / 256, 256, 0, stream>>>(ei, deg);
    k_dinv    <<<(N_NODES + 255) / 256, 256, 0, stream>>>(deg, dinv);

    k_gemm    <<<N_NODES / 32, 256, 0, stream>>>(x, W, h);           // 3125 blocks

    k_self    <<<(N_NODES * OUT_DIM + 255) / 256, 256, 0, stream>>>(h, dinv, b, out);

    // one wave (32 lanes) per edge
    long long scatter_threads = (long long)N_EDGES * 32;
    k_scatter <<<(unsigned)((scatter_threads + 255) / 256), 256, 0, stream>>>(ei, h, dinv, out);

    // one wave per row
    long long sm_threads = (long long)N_NODES * 32;
    k_logsoftmax<<<(unsigned)((sm_threads + 255) / 256), 256, 0, stream>>>(out);
}